// L1AttnSparse_38147899523350
// MI455X (gfx1250) — compile-verified
//
#include <hip/hip_runtime.h>
#include <hip/hip_bf16.h>
#include <stdint.h>
#include <stddef.h>

// ---------------------------------------------------------------------------
// L1 sparse attention for MI455X (gfx1250, wave32).
//   bs=2, n_tok=2048, n_heads=8, width=64, dst_mxlen=32
// One wave owns (b, h, 16 dst tokens). Phases (all intra-wave, no barriers):
//   P0: TDM tensor_load_to_lds stages the 16x64 fp32 q-tile into LDS,
//       then each lane hoists its token's q row into 16 float4 registers.
//   P1: lanes = (token, K-offset); each lane computes exp(-0.125*L1) for 2
//       slots per group -> values land directly in the WMMA f32 16x16x4
//       A-operand layout. 8x v_wmma_f32_16x16x4_f32 vs all-ones B produce all
//       16 softmax denominators on the matrix pipe. (e, src) pairs go to LDS
//       as one packed b64 store.
//   P2: lanes = width (float2); v_rcp_f32 reciprocal of the denominator,
//       coalesced v-row gathers + FMA, coalesced float2 stores of vo.
// ---------------------------------------------------------------------------

typedef float v2f __attribute__((ext_vector_type(2)));
typedef float v8f __attribute__((ext_vector_type(8)));
typedef unsigned int v4u __attribute__((ext_vector_type(4)));
typedef int v8i __attribute__((ext_vector_type(8)));
typedef int v4i __attribute__((ext_vector_type(4)));

#define BS    2
#define NTOK  2048
#define NH    8
#define WID   64
#define MX    32
#define SCALE (-0.125f)   // -1/sqrt(64)

struct SMem {
  float  qs[4][16][WID];    // 16 KB  (MUST stay first: TDM descriptor assumes LDS offset 0)
  float2 evsrc[4][16][MX];  // 16 KB  packed (exp value, __int_as_float(src))
  float  dn[4][16];         // 256 B  softmax denominators
};

__global__ __launch_bounds__(128) void l1attn_sparse_kernel(
    const float* __restrict__ vmat, const float* __restrict__ qmat,
    const float* __restrict__ kmat, const int* __restrict__ coo,
    float* __restrict__ out)
{
  __shared__ SMem sm;

  const int lane = threadIdx.x & 31;
  const int wv   = threadIdx.x >> 5;          // wave id 0..3
  const int bid  = blockIdx.x;
  const int tile = bid & 31;                  // 32 tiles of 64 tokens
  const int h    = (bid >> 5) & 7;
  const int b    = bid >> 8;
  const int tokBase = tile * 64 + wv * 16;    // this wave's 16 tokens

  // ---------------- Phase 0: stage q tile (16 tokens x 64 fp32) into LDS ----
#if __has_builtin(__builtin_amdgcn_tensor_load_to_lds)
  {
    const size_t qoff = (((size_t)b * NTOK + tokBase) * NH + h) * WID;
    unsigned long long ga = (unsigned long long)(uintptr_t)(qmat + qoff);
    const unsigned int ldsoff = (unsigned int)(wv * (16 * WID * 4));
    // D# group0: count=1 | lds_addr | global_addr[56:0] | type=2
    v4u g0 = { 1u, ldsoff, (unsigned int)ga,
               (unsigned int)((ga >> 32) & 0x1FFFFFFull) | (2u << 30) };
    // D# group1: data_size=4B, tensor_dim0=64, tensor_dim1=16,
    //            tile_dim0=64, tile_dim1=16, dim0_stride=NH*WID=512 elems
    v8i g1 = { 0x00020000,
               (WID << 16),       // tensor_dim0[15:0] at bits 63:48
               (16  << 16),       // tensor_dim1[15:0] at bits 95:80 (dim0 hi = 0)
               (WID << 16),       // tile_dim0 at bits 127:112 (dim1 hi = 0)
               16,                // tile_dim1 (tile_dim2 = 0)
               (NH * WID),        // tensor_dim0_stride = 512 elems
               0, 0 };
    v4i gz = { 0, 0, 0, 0 };
#if __has_include(<hip/amd_detail/amd_gfx1250_TDM.h>)
    v8i gz8 = { 0, 0, 0, 0, 0, 0, 0, 0 };     // therock clang-23: 6-arg form
    __builtin_amdgcn_tensor_load_to_lds(g0, g1, gz, gz, gz8, 0);
#else
    __builtin_amdgcn_tensor_load_to_lds(g0, g1, gz, gz, 0);  // ROCm 7.2 5-arg
#endif
  }
  __builtin_amdgcn_s_wait_tensorcnt(0);
#else
  {
    const float* qbase = qmat + (((size_t)b * NTOK + tokBase) * NH + h) * WID;
    float4* dst4 = (float4*)(&sm.qs[wv][0][0]);
    #pragma unroll
    for (int i = lane; i < 16 * 16; i += 32) {
      const int row = i >> 4, ch = i & 15;
      dst4[i] = ((const float4*)(qbase + (size_t)row * NH * WID))[ch];
    }
  }
#endif

  // ---------------- Phase 1: L1 distances + exp, denominators via WMMA -----
  // Lane layout matches V_WMMA_F32_16X16X4_F32 A-operand:
  //   m = lane&15 (M row = token), lanes 16..31 carry K+2.
  const int m    = lane & 15;
  const int koff = (lane >> 4) << 1;           // 0 or 2
  const int tok  = tokBase + m;

  // Hoist this lane's q row into registers: one pass of ds_load_b128, then
  // the distance loop is pure global k traffic + VALU.
  float4 qreg[16];
  {
    const float4* qsrow4 = (const float4*)(&sm.qs[wv][m][0]);
    #pragma unroll
    for (int wc = 0; wc < 16; ++wc) qreg[wc] = qsrow4[wc];
  }

  v8f c = {};
  const v2f ones = { 1.0f, 1.0f };
#if !__has_builtin(__builtin_amdgcn_wmma_f32_16x16x4_f32)
  float psum = 0.0f;
#endif

  for (int g = 0; g < 8; ++g) {
    float e2[2];
    #pragma unroll
    for (int j = 0; j < 2; ++j) {
      const int s   = 4 * g + koff + j;
      const int src = coo[(tok * MX + s) * 3 + 1];
      const float* krow = kmat + (((size_t)b * NTOK + src) * NH + h) * WID;
      float acc = 0.0f;
      #pragma unroll
      for (int wc = 0; wc < 16; ++wc) {
        const float4 kv = ((const float4*)krow)[wc];
        acc += __builtin_fabsf(qreg[wc].x - kv.x) + __builtin_fabsf(qreg[wc].y - kv.y)
             + __builtin_fabsf(qreg[wc].z - kv.z) + __builtin_fabsf(qreg[wc].w - kv.w);
      }
      const float e = __expf(acc * SCALE);
      sm.evsrc[wv][m][s] = make_float2(e, __int_as_float(src));  // one ds_store_b64
      e2[j] = e;
    }
#if __has_builtin(__builtin_amdgcn_wmma_f32_16x16x4_f32)
    v2f a = { e2[0], e2[1] };
    // D[m][*] += sum_k A[m,k]  ->  row sums (softmax denominators)
    c = __builtin_amdgcn_wmma_f32_16x16x4_f32(
        false, a, false, ones, (short)0, c, false, false);
#else
    psum += e2[0] + e2[1];
#endif
  }

#if __has_builtin(__builtin_amdgcn_wmma_f32_16x16x4_f32)
  // C/D layout: lane ln, VGPR r holds D[M = r + 8*(ln>=16)][N = ln&15];
  // every column equals the row sum, so lanes 0 and 16 publish all 16.
  if ((lane & 15) == 0) {
    const int base = (lane >> 4) * 8;
    #pragma unroll
    for (int r = 0; r < 8; ++r) sm.dn[wv][base + r] = c[r];
  }
#else
  {
    const float tot = psum + __shfl_xor(psum, 16);
    if (lane < 16) sm.dn[wv][lane] = tot;
  }
#endif

  // ---------------- Phase 2: normalized weighted-V gather ------------------
  // lanes = width (float2 each); v-row gathers are coalesced 256B/instr.
  const int w0 = lane << 1;
  for (int mm = 0; mm < 16; ++mm) {
    const float rd = __builtin_amdgcn_rcpf(sm.dn[wv][mm]);  // v_rcp_f32, no div expansion
    float ax = 0.0f, ay = 0.0f;
    #pragma unroll 4
    for (int s = 0; s < MX; ++s) {
      const float2 p  = sm.evsrc[wv][mm][s];                // one ds_load_b64
      const float wgt = p.x * rd;
      const int   src = __float_as_int(p.y);
      const float* vrow = vmat + (((size_t)b * NTOK + src) * NH + h) * WID;
      const float2 vv = *(const float2*)(vrow + w0);
      ax = fmaf(wgt, vv.x, ax);
      ay = fmaf(wgt, vv.y, ay);
    }
    if (mm + 1 < 16) {
      const int nsrc = __float_as_int(sm.evsrc[wv][mm + 1][0].y);
      __builtin_prefetch(vmat + (((size_t)b * NTOK + nsrc) * NH + h) * WID + w0, 0, 0);
    }
    float* orow = out + (((size_t)b * NTOK + (tokBase + mm)) * NH + h) * WID;
    *((float2*)(orow + w0)) = make_float2(ax, ay);
  }
}

extern "C" void kernel_launch(void* const* d_in, const int* in_sizes, int n_in,
                              void* d_out, int out_size, void* d_ws, size_t ws_size,
                              hipStream_t stream) {
  (void)in_sizes; (void)n_in; (void)out_size; (void)d_ws; (void)ws_size;
  const float* v   = (const float*)d_in[0];
  const float* q   = (const float*)d_in[1];
  const float* k   = (const float*)d_in[2];
  const int*   coo = (const int*)d_in[3];
  float* out = (float*)d_out;

  dim3 grid(BS * NH * (NTOK / 64));   // 512 blocks
  dim3 block(128);                    // 4 waves of 32
  hipLaunchKernelGGL(l1attn_sparse_kernel, grid, block, 0, stream,
                     v, q, k, coo, out);
}